// StackedTriConv_74586402063281
// MI455X (gfx1250) — compile-verified
//
#include <hip/hip_runtime.h>
#include <hip/hip_bf16.h>

typedef __attribute__((ext_vector_type(2))) float v2f;
typedef __attribute__((ext_vector_type(8))) float v8f;

// ---------------------------------------------------------------------------
// Utility kernels
// ---------------------------------------------------------------------------
__global__ void zero_kernel(float* __restrict__ p, long long n) {
  long long i = (long long)blockIdx.x * blockDim.x + threadIdx.x;
  if (i < n) p[i] = 0.0f;
}

__global__ void degree_kernel(const int* __restrict__ dst, float* __restrict__ deg, int E) {
  int e = blockIdx.x * blockDim.x + threadIdx.x;
  if (e < E) atomicAdd(&deg[dst[e]], 1.0f);
}

// Wc = Wt @ Wp  (Wt: [CIN, COUT], Wp: [COUT, COUT]),  btp = bt @ Wp
template <int CIN, int COUT>
__global__ void combine_kernel(const float* __restrict__ Wt, const float* __restrict__ bt,
                               const float* __restrict__ Wp, float* __restrict__ Wc,
                               float* __restrict__ btp) {
  int t = blockIdx.x * blockDim.x + threadIdx.x;
  const int total = (CIN + 1) * COUT;
  if (t >= total) return;
  int i = t / COUT;
  int j = t - i * COUT;
  float acc = 0.0f;
  if (i < CIN) {
    for (int k = 0; k < COUT; ++k) acc += Wt[i * COUT + k] * Wp[k * COUT + j];
    Wc[i * COUT + j] = acc;
  } else {
    for (int k = 0; k < COUT; ++k) acc += bt[k] * Wp[k * COUT + j];
    btp[j] = acc;
  }
}

// ---------------------------------------------------------------------------
// Neighbor scatter-sum: s[dst[e]][c] += x[src[e]][c]
// One 2^LOG-thread group per edge -> coalesced gather, contiguous atomics.
// Node tables (~19MB) live in the 192MB L2, so atomics never touch HBM.
// ---------------------------------------------------------------------------
template <int C, int LOG>
__global__ void scatter_kernel(const float* __restrict__ x, const int* __restrict__ src,
                               const int* __restrict__ dst, float* __restrict__ s, int E) {
  long long t = (long long)blockIdx.x * blockDim.x + threadIdx.x;
  int e = (int)(t >> LOG);
  int c = (int)(t & ((1 << LOG) - 1));
  if (e < E && c < C) {
    atomicAdd(&s[(long long)dst[e] * C + c], x[(long long)src[e] * C + c]);
  }
}

// ---------------------------------------------------------------------------
// Fused TriConv layer GEMM (full fp32 via V_WMMA_F32_16X16X4_F32):
//   out[r][c] = act( sum_k (deg[r]*x[r][k] - s[r][k]) * Wc[k][c]
//                    + deg[r]*btp[c] + bp[c] )
// One wave -> 16 rows x COUT cols.  Block = 256 threads = 8 waves = 128 rows.
//
// Fragment layouts (CDNA5 ISA 7.12.2, 32-bit, wave32):
//   A 16x4 : lanes 0-15 hold M=lane, {v0,v1} = K0,K1 ; lanes 16-31 = K2,K3
//   B 4x16 : lanes 0-15 hold N=lane, {v0,v1} = K0,K1 ; lanes 16-31 = K2,K3
//   C/D    : VGPR v, lane L -> row = v + 8*(L>>4), col = L&15
// ---------------------------------------------------------------------------
template <int CIN, int COUT, bool RELU>
__global__ void triconv_gemm_kernel(const float* __restrict__ x, const float* __restrict__ s,
                                    const float* __restrict__ deg, const float* __restrict__ Wc,
                                    const float* __restrict__ btp, const float* __restrict__ bp,
                                    float* __restrict__ out, int n) {
  constexpr int NT = COUT / 16;
  const int lane = threadIdx.x & 31;
  const int wave = threadIdx.x >> 5;
  const int rowBase = (blockIdx.x * (int)(blockDim.x >> 5) + wave) * 16;
  if (rowBase >= n) return;  // wave-uniform: EXEC stays all-ones for WMMA

  const int col = lane & 15;
  const int khalf = (lane >> 4) << 1;  // 0 or 2
  int arow = rowBase + col;
  if (arow >= n) arow = n - 1;  // clamp (n is a multiple of 16 in practice)
  const float dga = deg[arow];

  v8f acc[NT] = {};

#pragma unroll
  for (int k0 = 0; k0 < CIN; k0 += 4) {
    const int ka = k0 + khalf;
    v2f a;
    a.x = dga * x[(long long)arow * CIN + ka] - s[(long long)arow * CIN + ka];
    a.y = dga * x[(long long)arow * CIN + ka + 1] - s[(long long)arow * CIN + ka + 1];
#pragma unroll
    for (int t = 0; t < NT; ++t) {
      v2f b;
      b.x = Wc[ka * COUT + t * 16 + col];
      b.y = Wc[(ka + 1) * COUT + t * 16 + col];
      acc[t] = __builtin_amdgcn_wmma_f32_16x16x4_f32(false, a, false, b, (short)0, acc[t],
                                                     false, false);
    }
  }

  // Epilogue: per-row bias deg[row]*btp[c] + bp[c], optional ReLU.
#pragma unroll
  for (int t = 0; t < NT; ++t) {
    const int c = t * 16 + col;
    const float bpc = bp[c];
    const float btc = btp[c];
#pragma unroll
    for (int v = 0; v < 8; ++v) {
      const int row = rowBase + v + ((lane >> 4) << 3);
      if (row < n) {
        float val = acc[t][v] + deg[row] * btc + bpc;
        if (RELU) val = val > 0.0f ? val : 0.0f;
        out[(long long)row * COUT + c] = val;
      }
    }
  }
}

// ---------------------------------------------------------------------------
// In-place row softmax over 32 columns: one wave32 per row.
// ---------------------------------------------------------------------------
__global__ void softmax32_kernel(float* __restrict__ out, int n) {
  const int row = blockIdx.x * (int)(blockDim.x >> 5) + (threadIdx.x >> 5);
  const int lane = threadIdx.x & 31;
  if (row >= n) return;
  float v = out[(long long)row * 32 + lane];
  float m = v;
#pragma unroll
  for (int off = 16; off > 0; off >>= 1) {
    float o = __shfl_xor(m, off, 32);
    m = m > o ? m : o;
  }
  float e = __expf(v - m);
  float ssum = e;
#pragma unroll
  for (int off = 16; off > 0; off >>= 1) ssum += __shfl_xor(ssum, off, 32);
  out[(long long)row * 32 + lane] = e / ssum;
}

// ---------------------------------------------------------------------------
static inline unsigned cdiv_ll(long long a, long long b) { return (unsigned)((a + b - 1) / b); }

extern "C" void kernel_launch(void* const* d_in, const int* in_sizes, int n_in,
                              void* d_out, int out_size, void* d_ws, size_t ws_size,
                              hipStream_t stream) {
  const float* f   = (const float*)d_in[0];
  const int*   ei  = (const int*)d_in[1];
  const float* Wt1 = (const float*)d_in[2];  const float* bt1 = (const float*)d_in[3];
  const float* Wp1 = (const float*)d_in[4];  const float* bp1 = (const float*)d_in[5];
  const float* Wt2 = (const float*)d_in[6];  const float* bt2 = (const float*)d_in[7];
  const float* Wp2 = (const float*)d_in[8];  const float* bp2 = (const float*)d_in[9];
  const float* Wt3 = (const float*)d_in[10]; const float* bt3 = (const float*)d_in[11];
  const float* Wp3 = (const float*)d_in[12]; const float* bp3 = (const float*)d_in[13];

  const int N = in_sizes[0] / 32;
  const int E = in_sizes[1] / 2;
  const int* src = ei;
  const int* dst = ei + E;

  // Workspace carve-up
  float* ws   = (float*)d_ws;
  float* deg  = ws;  ws += N;
  float* sbuf = ws;  ws += (size_t)N * 48;
  float* h1   = ws;  ws += (size_t)N * 48;
  float* h2   = ws;  ws += (size_t)N * 48;
  float* Wc1  = ws;  ws += 32 * 48;
  float* btp1 = ws;  ws += 48;
  float* Wc2  = ws;  ws += 48 * 48;
  float* btp2 = ws;  ws += 48;
  float* Wc3  = ws;  ws += 48 * 32;
  float* btp3 = ws;  ws += 32;
  float* outp = (float*)d_out;

  const int TB = 256;

  // Precompute collapsed weights Wc_l = Wt_l @ Wp_l, btp_l = bt_l @ Wp_l (tiny)
  combine_kernel<32, 48><<<cdiv_ll(33 * 48, TB), TB, 0, stream>>>(Wt1, bt1, Wp1, Wc1, btp1);
  combine_kernel<48, 48><<<cdiv_ll(49 * 48, TB), TB, 0, stream>>>(Wt2, bt2, Wp2, Wc2, btp2);
  combine_kernel<48, 32><<<cdiv_ll(49 * 32, TB), TB, 0, stream>>>(Wt3, bt3, Wp3, Wc3, btp3);

  // In-degree
  zero_kernel<<<cdiv_ll(N, TB), TB, 0, stream>>>(deg, N);
  degree_kernel<<<cdiv_ll(E, TB), TB, 0, stream>>>(dst, deg, E);

  const unsigned gemmBlocks = cdiv_ll(N, 128);  // 8 waves * 16 rows per block

  // Layer 1: 32 -> 48, ReLU
  zero_kernel<<<cdiv_ll((long long)N * 32, TB), TB, 0, stream>>>(sbuf, (long long)N * 32);
  scatter_kernel<32, 5><<<cdiv_ll((long long)E << 5, TB), TB, 0, stream>>>(f, src, dst, sbuf, E);
  triconv_gemm_kernel<32, 48, true><<<gemmBlocks, TB, 0, stream>>>(f, sbuf, deg, Wc1, btp1, bp1,
                                                                   h1, N);

  // Layer 2: 48 -> 48, ReLU
  zero_kernel<<<cdiv_ll((long long)N * 48, TB), TB, 0, stream>>>(sbuf, (long long)N * 48);
  scatter_kernel<48, 6><<<cdiv_ll((long long)E << 6, TB), TB, 0, stream>>>(h1, src, dst, sbuf, E);
  triconv_gemm_kernel<48, 48, true><<<gemmBlocks, TB, 0, stream>>>(h1, sbuf, deg, Wc2, btp2, bp2,
                                                                   h2, N);

  // Layer 3: 48 -> 32, logits straight into d_out, then in-place softmax
  zero_kernel<<<cdiv_ll((long long)N * 48, TB), TB, 0, stream>>>(sbuf, (long long)N * 48);
  scatter_kernel<48, 6><<<cdiv_ll((long long)E << 6, TB), TB, 0, stream>>>(h2, src, dst, sbuf, E);
  triconv_gemm_kernel<48, 32, false><<<gemmBlocks, TB, 0, stream>>>(h2, sbuf, deg, Wc3, btp3, bp3,
                                                                    outp, N);
  softmax32_kernel<<<cdiv_ll(N, 8), TB, 0, stream>>>(outp, N);
}